// OnlineTripletLoss_40733469835388
// MI455X (gfx1250) — compile-verified
//
#include <hip/hip_runtime.h>
#include <hip/hip_bf16.h>

#define B 256
#define DIM 512
#define MARGIN 0.2f
#define TOPK 4096u

typedef __attribute__((ext_vector_type(2))) float v2f;
typedef __attribute__((ext_vector_type(8))) float v8f;

// ---------------------------------------------------------------------------
// Kernel 1: row squared norms. 256 blocks (one per row), 256 threads.
// ---------------------------------------------------------------------------
__global__ void otl_row_norms(const float* __restrict__ X, float* __restrict__ nx) {
    const int i = blockIdx.x;
    const int tid = threadIdx.x;
    float a = X[i * DIM + tid];
    float b = X[i * DIM + tid + 256];
    __shared__ float red[256];
    red[tid] = a * a + b * b;
    __syncthreads();
    for (int off = 128; off > 0; off >>= 1) {
        if (tid < off) red[tid] += red[tid + off];
        __syncthreads();
    }
    if (tid == 0) nx[i] = red[0];
}

// ---------------------------------------------------------------------------
// Kernel 2: D[i,j] = nx[i] + nx[j] - 2 * (X X^T)[i,j] via V_WMMA_F32_16X16X4_F32.
// Grid: 16x16 tiles, one wave (32 threads) per 16x16 output tile.
// A (16x4): lanes 0-15 -> M, VGPR0/1 = K 0/1; lanes 16-31 -> K 2/3.
// B (4x16): lanes 0-15 -> N, VGPR0/1 = K 0/1; lanes 16-31 -> K 2/3.
// C/D (16x16 f32): lane<16: VGPRv = (M=v, N=lane); lane>=16: (M=v+8, N=lane-16).
// ---------------------------------------------------------------------------
__global__ void otl_gram_wmma(const float* __restrict__ X,
                              const float* __restrict__ nx,
                              float* __restrict__ D) {
    const int ti = blockIdx.x;          // row tile
    const int tj = blockIdx.y;          // col tile
    const int lane = threadIdx.x;       // 0..31
    const int half = lane >> 4;         // 0 or 1
    const int lm = lane & 15;

    const int row = ti * 16 + lm;       // M index carried by this lane (A)
    const int col = tj * 16 + lm;       // N index carried by this lane (B)

    v8f acc = {};
    for (int k = 0; k < DIM; k += 4) {
        const int ka = k + half * 2;    // this half-wave holds K = ka, ka+1
        v2f a, b;
        a[0] = X[row * DIM + ka];
        a[1] = X[row * DIM + ka + 1];
        b[0] = X[col * DIM + ka];       // B[k][n] = X[n][k]  (Gram: B = X^T tile)
        b[1] = X[col * DIM + ka + 1];
        acc = __builtin_amdgcn_wmma_f32_16x16x4_f32(
            /*neg_a=*/false, a, /*neg_b=*/false, b,
            /*c_mod=*/(short)0, acc, /*reuse_a=*/false, /*reuse_b=*/false);
    }

    const int n_ = lm;
#pragma unroll
    for (int v = 0; v < 8; ++v) {
        const int m_ = v + half * 8;
        const int gi = ti * 16 + m_;
        const int gj = tj * 16 + n_;
        D[gi * B + gj] = nx[gi] + nx[gj] - 2.0f * acc[v];
    }
}

// ---------------------------------------------------------------------------
// Kernel 3: init radix-select state.
// ---------------------------------------------------------------------------
__global__ void otl_sel_init(unsigned* __restrict__ hist, unsigned* __restrict__ state) {
    hist[threadIdx.x] = 0u;
    if (threadIdx.x == 0) {
        state[0] = 0u;      // prefix bits fixed so far
        state[1] = TOPK;    // k remaining within current prefix
    }
}

// ---------------------------------------------------------------------------
// Kernel 4 (x4): histogram pass over valid triplets. One block per q,
// threads over n, loop over valid p (p<q, label match). Loss >= 0 always,
// so f32 bit pattern is monotone as u32.
// ---------------------------------------------------------------------------
__global__ void otl_hist_pass(const float* __restrict__ D,
                              const long long* __restrict__ t,
                              const unsigned* __restrict__ state,
                              unsigned* __restrict__ hist,
                              int shift) {
    const int q = blockIdx.x;
    const int n = threadIdx.x;

    __shared__ float drow[B];
    __shared__ int   lt[B];
    __shared__ unsigned lhist[256];

    drow[n]  = D[q * B + n];
    lt[n]    = (int)t[n];
    lhist[n] = 0u;
    __syncthreads();

    const int tq = lt[q];
    const unsigned prefix = state[0];
    const unsigned maskhi = (shift == 24) ? 0u : (0xFFFFFFFFu << (shift + 8));
    const float dqn = drow[n];

    if (lt[n] != tq) {                      // valid negative
        for (int p = 0; p < q; ++p) {
            if (lt[p] != tq) continue;      // valid positive (p<q, same label)
            float loss = drow[p] - dqn + MARGIN;
            loss = fmaxf(loss, 0.0f);
            const unsigned bits = __float_as_uint(loss);
            if ((bits & maskhi) == (prefix & maskhi))
                atomicAdd(&lhist[(bits >> shift) & 0xFFu], 1u);
        }
    }
    __syncthreads();
    if (lhist[n]) atomicAdd(&hist[n], lhist[n]);
}

// ---------------------------------------------------------------------------
// Kernel 5 (x4): pick digit from histogram (scan from top), update state,
// zero histogram for next pass.
// ---------------------------------------------------------------------------
__global__ void otl_select_digit(unsigned* __restrict__ hist,
                                 unsigned* __restrict__ state,
                                 int shift) {
    if (threadIdx.x == 0) {
        const unsigned k = state[1];
        unsigned acc = 0u;
        int b = 255;
        for (; b >= 0; --b) {
            const unsigned c = hist[b];
            if (acc + c >= k) break;
            acc += c;
        }
        if (b < 0) b = 0;  // degenerate safety
        state[0] |= ((unsigned)b) << shift;
        state[1] = k - acc;
    }
    __syncthreads();
    hist[threadIdx.x] = 0u;
}

// ---------------------------------------------------------------------------
// Kernel 6: final pass. Sum/count of values strictly greater than kth value.
// Per-q partials via deterministic LDS tree reduction; integer + fixed-order
// float adds only (bit-deterministic, graph-replay safe).
// ---------------------------------------------------------------------------
__global__ void otl_final_pass(const float* __restrict__ D,
                               const long long* __restrict__ t,
                               const unsigned* __restrict__ state,
                               float* __restrict__ psum,
                               unsigned* __restrict__ pcnt) {
    const int q = blockIdx.x;
    const int n = threadIdx.x;

    __shared__ float drow[B];
    __shared__ int   lt[B];
    __shared__ float ssum[256];
    __shared__ unsigned scnt[256];

    drow[n] = D[q * B + n];
    lt[n]   = (int)t[n];
    __syncthreads();

    const int tq = lt[q];
    const unsigned kbits = state[0];
    const float dqn = drow[n];

    float lsum = 0.0f;
    unsigned lcnt = 0u;
    if (lt[n] != tq) {
        for (int p = 0; p < q; ++p) {
            if (lt[p] != tq) continue;
            float loss = drow[p] - dqn + MARGIN;
            loss = fmaxf(loss, 0.0f);
            if (__float_as_uint(loss) > kbits) { lsum += loss; ++lcnt; }
        }
    }
    ssum[n] = lsum;
    scnt[n] = lcnt;
    __syncthreads();
    for (int off = 128; off > 0; off >>= 1) {
        if (n < off) { ssum[n] += ssum[n + off]; scnt[n] += scnt[n + off]; }
        __syncthreads();
    }
    if (n == 0) { psum[q] = ssum[0]; pcnt[q] = scnt[0]; }
}

// ---------------------------------------------------------------------------
// Kernel 7: finalize. mean = (sum_gt + (K - count_gt) * kth) / K.
// ---------------------------------------------------------------------------
__global__ void otl_finalize(const float* __restrict__ psum,
                             const unsigned* __restrict__ pcnt,
                             const unsigned* __restrict__ state,
                             float* __restrict__ out) {
    const int n = threadIdx.x;
    __shared__ float ssum[256];
    __shared__ unsigned scnt[256];
    ssum[n] = psum[n];
    scnt[n] = pcnt[n];
    __syncthreads();
    for (int off = 128; off > 0; off >>= 1) {
        if (n < off) { ssum[n] += ssum[n + off]; scnt[n] += scnt[n + off]; }
        __syncthreads();
    }
    if (n == 0) {
        const float kth = __uint_as_float(state[0]);
        unsigned cnt = scnt[0];
        if (cnt > TOPK) cnt = TOPK;  // safety clamp
        const float mean = (ssum[0] + (float)(TOPK - cnt) * kth) / (float)TOPK;
        out[0] = mean;
    }
}

// ---------------------------------------------------------------------------
// Launch: norms -> WMMA Gram/distances -> radix-select (4 passes) -> final.
// Workspace layout (bytes):
//   [0,       262144) D matrix (256x256 f32)
//   [262144,  263168) row norms (256 f32)
//   [263168,  264192) histogram (256 u32)
//   [264192,  264208) select state (prefix, k_remaining)
//   [264208,  265232) per-q partial sums (256 f32)
//   [265232,  266256) per-q partial counts (256 u32)
// ---------------------------------------------------------------------------
extern "C" void kernel_launch(void* const* d_in, const int* in_sizes, int n_in,
                              void* d_out, int out_size, void* d_ws, size_t ws_size,
                              hipStream_t stream) {
    const float* X = (const float*)d_in[0];
    const long long* t = (const long long*)d_in[1];
    float* out = (float*)d_out;

    char* ws = (char*)d_ws;
    float*    Dm    = (float*)(ws + 0);
    float*    nx    = (float*)(ws + 262144);
    unsigned* hist  = (unsigned*)(ws + 263168);
    unsigned* state = (unsigned*)(ws + 264192);
    float*    psum  = (float*)(ws + 264208);
    unsigned* pcnt  = (unsigned*)(ws + 265232);

    otl_row_norms<<<B, 256, 0, stream>>>(X, nx);
    otl_gram_wmma<<<dim3(B / 16, B / 16), 32, 0, stream>>>(X, nx, Dm);
    otl_sel_init<<<1, 256, 0, stream>>>(hist, state);

    for (int shift = 24; shift >= 0; shift -= 8) {
        otl_hist_pass<<<B, 256, 0, stream>>>(Dm, t, state, hist, shift);
        otl_select_digit<<<1, 256, 0, stream>>>(hist, state, shift);
    }

    otl_final_pass<<<B, 256, 0, stream>>>(Dm, t, state, psum, pcnt);
    otl_finalize<<<1, 256, 0, stream>>>(psum, pcnt, state, out);
}